// RevEcho_65403761983945
// MI455X (gfx1250) — compile-verified
//
#include <hip/hip_runtime.h>
#include <stdint.h>

#define T_LEN   160000
#define BCH     16
#define MAX_E   512                       // static cap; max_echoes <= ~147 by construction
#define SEGS    6                         // (source 0/1) x (repeat 0..2)
#define CNT_W   8                         // 8 dwords of per-segment counts (6 used)
#define TAB_WORDS (CNT_W + SEGS * MAX_E * 2)   // 6152 dwords = 24608 B (counts + interleaved taps)

struct U2  { uint32_t x, y; };
struct Tap { int32_t d; float g; };       // 8 bytes, LDS ds_load_b64 friendly

// ---------------------------------------------------------------------------
// Exact JAX Threefry-2x32 (20 rounds), matching jax._src.prng
// ---------------------------------------------------------------------------
__device__ __forceinline__ U2 threefry2x32(uint32_t k0, uint32_t k1,
                                           uint32_t x0, uint32_t x1) {
  const uint32_t k2 = k0 ^ k1 ^ 0x1BD11BDAu;
  x0 += k0; x1 += k1;
#define TF_ROUND(r) { x0 += x1; x1 = (x1 << (r)) | (x1 >> (32 - (r))); x1 ^= x0; }
  TF_ROUND(13) TF_ROUND(15) TF_ROUND(26) TF_ROUND(6)
  x0 += k1; x1 += k2 + 1u;
  TF_ROUND(17) TF_ROUND(29) TF_ROUND(16) TF_ROUND(24)
  x0 += k2; x1 += k0 + 2u;
  TF_ROUND(13) TF_ROUND(15) TF_ROUND(26) TF_ROUND(6)
  x0 += k0; x1 += k1 + 3u;
  TF_ROUND(17) TF_ROUND(29) TF_ROUND(16) TF_ROUND(24)
  x0 += k1; x1 += k2 + 4u;
  TF_ROUND(13) TF_ROUND(15) TF_ROUND(26) TF_ROUND(6)
  x0 += k2; x1 += k0 + 5u;
#undef TF_ROUND
  return {x0, x1};
}

// jax.random.split(key, N): counts=iota(2N); pairs (j, N+j); flat=[X0s..,X1s..];
// keys[i] = (flat[2i], flat[2i+1]).  split_word(j) returns flat[j].
__device__ __forceinline__ uint32_t split_word(U2 key, int j, int N) {
  const int jj = (j < N) ? j : j - N;
  U2 e = threefry2x32(key.x, key.y, (uint32_t)jj, (uint32_t)(jj + N));
  return (j < N) ? e.x : e.y;
}

// jax.random.uniform(k, (), f32, -1, 1): bits = enc_k(0,0).x
__device__ __forceinline__ float jax_uniform_pm1(uint32_t k0, uint32_t k1) {
  const uint32_t bits = threefry2x32(k0, k1, 0u, 0u).x;
  const float u = __uint_as_float((bits >> 9) | 0x3f800000u) - 1.0f;  // [0,1)
  const float r = fmaf(u, 2.0f, -1.0f);                               // *(hi-lo)+lo
  return fmaxf(-1.0f, r);                                             // lax.max(minval, .)
}

// ---------------------------------------------------------------------------
// Kernel A: reproduce the PRNG/scan scalar chain; emit (cumulative delay, gain)
// tap tables per (source, repeat) into d_ws.
// ws layout (dwords): [0..7] counts | [8 ..] Tap[SEGS][MAX_E] interleaved {d,g}
// ---------------------------------------------------------------------------
__global__ void revecho_prep(const float* __restrict__ pInitial,
                             const float* __restrict__ pFirstDelay,
                             const float* __restrict__ pRt60,
                             const int*   __restrict__ pMaxEchoes,
                             uint32_t*    __restrict__ ws) {
  const int tid = threadIdx.x;
  if (tid >= SEGS) { if (tid < CNT_W) ws[tid] = 0u; return; }

  const float initial = *pInitial;
  const float fd      = *pFirstDelay;
  const float rt60    = *pRt60;
  const int   N       = *pMaxEchoes;

  const int s = tid / 3;   // 0 = noise, 1 = clean
  const int r = tid % 3;   // repeat

  // key = jax.random.key(42) = (0,42); fold_in(key, s); fold_in(that, r)
  U2 base = threefry2x32(0u, 42u, 0u, (uint32_t)s);
  U2 rep  = threefry2x32(base.x, base.y, 0u, (uint32_t)r);

  Tap* tp = (Tap*)(ws + CNT_W) + tid * MAX_E;

  float frac = 1.0f;
  int   Dcum = 0;
  int   cnt  = 0;

  for (int n = 0; n < N && cnt < MAX_E; ++n) {
    if (!(frac > 0.001f)) break;                    // scan goes inert from here on
    // keys[n] from split(rep, N)
    const uint32_t w0 = split_word(rep, 2 * n, N);
    const uint32_t w1 = split_word(rep, 2 * n + 1, N);
    // k1,k2 = split(keys[n]) : pairs (0,2),(1,3)
    U2 e0 = threefry2x32(w0, w1, 0u, 2u);
    U2 e1 = threefry2x32(w0, w1, 1u, 3u);
    // k1=(e0.x,e1.x), k2=(e0.y,e1.y)
    const float j1 = 1.0f + 0.1f * jax_uniform_pm1(e0.x, e1.x);
    int delay = 1 + (int)floorf((j1 * fd) * 16000.0f);
    delay = (delay < T_LEN) ? delay : T_LEN;

    Dcum += delay;
    if (Dcum >= T_LEN) break;                       // all further taps shift past the end
    tp[cnt].d = Dcum;
    tp[cnt].g = initial * frac;                     // gain uses frac BEFORE this step's att
    ++cnt;

    const float j2  = 1.0f + 0.1f * jax_uniform_pm1(e0.y, e1.y);
    const float att = exp10f(((-3.0f * j2) * fd) / rt60);
    frac *= att;
  }
  ws[tid] = (uint32_t)cnt;
}

// ---------------------------------------------------------------------------
// Kernel B: sparse-FIR gather-accumulate. Tap tables staged into LDS via the
// Tensor Data Mover (CDNA5 TDM) when available.
// ---------------------------------------------------------------------------
#if defined(__gfx1250__) && __has_builtin(__builtin_amdgcn_tensor_load_to_lds)
#define REVECHO_HAVE_TDM 1
typedef unsigned int revecho_v4u __attribute__((ext_vector_type(4)));
typedef int          revecho_v8i __attribute__((ext_vector_type(8)));
typedef int          revecho_v4i __attribute__((ext_vector_type(4)));
#endif

// Taps within a segment have strictly increasing cumulative delay d.
// Fast path (no clamps/selects) when the whole block tile is past every tap
// and fully in-bounds; guarded path (with monotonic early-break) otherwise.
__device__ __forceinline__ void accum_taps(const float* __restrict__ row,
                                           const Tap* __restrict__ taps,
                                           const int* __restrict__ cnts,
                                           int seg0, int tb, int blockMin,
                                           bool inBounds, float acc[4]) {
  for (int seg = seg0; seg < seg0 + 3; ++seg) {
    const int  cnt = cnts[seg];
    if (cnt <= 0) continue;
    const Tap* tp   = taps + seg * MAX_E;
    const int  maxD = tp[cnt - 1].d;                 // wave-uniform LDS read

    if (inBounds && blockMin >= maxD) {
      // ---- fast path: every idx in [0, T) for every lane/k ----
      const float* basep = row + tb;
      for (int i = 0; i < cnt; ++i) {
        const Tap t = tp[i];                         // ds_load_b64, wave-uniform
        const float* p = basep - t.d;
#pragma unroll
        for (int k = 0; k < 4; ++k)                  // imm-offset b32 loads
          acc[k] = fmaf(t.g, p[k << 8], acc[k]);
      }
    } else {
      // ---- boundary path ----
      const int tmax = blockMin + 1023;              // largest t this block touches
      for (int i = 0; i < cnt; ++i) {
        const Tap t = tp[i];
        if (t.d > tmax) break;                       // monotonic: rest contribute 0
        const int base = tb - t.d;
#pragma unroll
        for (int k = 0; k < 4; ++k) {
          int idx  = base + (k << 8);
          int idxc = (idx < 0) ? 0 : idx;
          idxc     = (idxc >= T_LEN) ? (T_LEN - 1) : idxc;   // memory-safe clamp
          const float v = row[idxc];
          acc[k] += (idx >= 0) ? t.g * v : 0.0f;
        }
      }
    }
  }
}

__global__ void __launch_bounds__(256)
revecho_main(const float* __restrict__ wav, float* __restrict__ out,
             const uint32_t* __restrict__ ws) {
  __shared__ alignas(16) uint32_t sTab[TAB_WORDS];
  const int tid = threadIdx.x;

#if defined(REVECHO_HAVE_TDM)
  if (tid < 32) {  // wave 0 issues one TDM DMA: ws[0..TAB_WORDS) -> LDS
    const uint64_t ga  = (uint64_t)(uintptr_t)ws;
    const uint32_t lds = (uint32_t)(uintptr_t)&sTab[0];   // low 32 bits = LDS offset
    const uint32_t N   = (uint32_t)TAB_WORDS;             // elements of 4 bytes
    // D# group0: count=1 | lds_addr | global_addr[56:0] | type=2
    revecho_v4u g0 = {1u, lds, (uint32_t)ga,
                      (uint32_t)((ga >> 32) & 0x01FFFFFFu) | (2u << 30)};
    // D# group1: data_size=4B; tensor_dim0=N, tensor_dim1=1; tile 1xN; stride0=N
    revecho_v8i g1 = {(int)0x00020000u,
                      (int)((N & 0xFFFFu) << 16),
                      (int)((N >> 16) | (1u << 16)),
                      (int)((N & 0xFFFFu) << 16),
                      1, (int)N, 0, 0};
    revecho_v4i g2 = {0, 0, 0, 0};          // groups 2/3 unused for a 2D tile
    revecho_v4i g3 = {0, 0, 0, 0};
    revecho_v8i g4 = {0, 0, 0, 0, 0, 0, 0, 0};  // trailing group (clang-23 6-arg form)
    __builtin_amdgcn_tensor_load_to_lds(g0, g1, g2, g3, g4, 0);
#if __has_builtin(__builtin_amdgcn_s_wait_tensorcnt)
    __builtin_amdgcn_s_wait_tensorcnt(0);
#else
    asm volatile("s_wait_tensorcnt 0x0" ::: "memory");
#endif
  }
#else
  for (int i = tid; i < TAB_WORDS; i += 256) sTab[i] = ws[i];
#endif
  __syncthreads();

  const int*  sCnt = (const int*)sTab;
  const Tap*  taps = (const Tap*)(sTab + CNT_W);

  const int b        = blockIdx.y;
  const int blockMin = blockIdx.x * 1024;
  const int tb       = blockMin + tid;             // 4 samples/thread, stride 256
  const bool inBounds = (blockMin + 1023) < T_LEN; // uniform: full block in range

  const float* nrow = wav + (size_t)b * T_LEN;             // noise[b]
  const float* crow = wav + (size_t)(BCH + b) * T_LEN;     // clean[b]
  __builtin_prefetch(nrow + tb, 0, 1);                     // global_prefetch_b8
  __builtin_prefetch(crow + tb, 0, 1);

  float aN[4] = {0.f, 0.f, 0.f, 0.f};
  float aC[4] = {0.f, 0.f, 0.f, 0.f};
  accum_taps(nrow, taps, sCnt, 0, tb, blockMin, inBounds, aN);  // reverb(noise)
  accum_taps(crow, taps, sCnt, 3, tb, blockMin, inBounds, aC);  // reverb(clean)

  float* outN = out;
  float* outC = out + (size_t)BCH * T_LEN;
  const size_t rowoff = (size_t)b * T_LEN;
#pragma unroll
  for (int k = 0; k < 4; ++k) {
    const int t = tb + (k << 8);
    if (t < T_LEN) {
      const float nv = nrow[t];
      const float cv = crow[t];
      __builtin_nontemporal_store(nv + aN[k] + 0.9f * aC[k], &outN[rowoff + t]);
      __builtin_nontemporal_store(cv + 0.1f * aC[k],         &outC[rowoff + t]);
    }
  }
}

// ---------------------------------------------------------------------------
extern "C" void kernel_launch(void* const* d_in, const int* in_sizes, int n_in,
                              void* d_out, int out_size, void* d_ws, size_t ws_size,
                              hipStream_t stream) {
  (void)in_sizes; (void)n_in; (void)out_size; (void)ws_size;
  const float* wav     = (const float*)d_in[0];   // [2,16,160000,1] f32
  const float* initial = (const float*)d_in[1];
  const float* fdly    = (const float*)d_in[2];
  const float* rt60    = (const float*)d_in[3];
  const int*   maxe    = (const int*)d_in[4];
  float*       out     = (float*)d_out;
  uint32_t*    ws      = (uint32_t*)d_ws;

  revecho_prep<<<dim3(1), dim3(32), 0, stream>>>(initial, fdly, rt60, maxe, ws);

  dim3 grid((T_LEN + 1023) / 1024, BCH);
  revecho_main<<<grid, dim3(256), 0, stream>>>(wav, out, ws);
}